// KohnShamSolver_62758062129416
// MI455X (gfx1250) — compile-verified
//
#include <hip/hip_runtime.h>
#include <hip/hip_bf16.h>

#define NX 4096

typedef __attribute__((ext_vector_type(2))) float v2f;
typedef __attribute__((ext_vector_type(8))) float v8f;

// ---------------------------------------------------------------------------
// Kernel 1: build Hamiltonian diagonal d[i] and Gershgorin spectral bounds.
//   d[i] = 1/dx^2 + V_ext + V_H + V_xc + l(l+1)/(2 r^2)
//   off-diagonal c = -0.5/dx^2 (constant)
// Single block, LDS min/max reduction.
// ---------------------------------------------------------------------------
__global__ void ks_build_diag(const int* __restrict__ lptr,
                              const float* __restrict__ V_ext,
                              const float* __restrict__ V_H,
                              const float* __restrict__ V_xc,
                              const float* __restrict__ xc,
                              const float* __restrict__ dxp,
                              float* __restrict__ diag,
                              float* __restrict__ bounds) {
    __shared__ float smin[1024];
    __shared__ float smax[1024];
    const float dx = dxp[0];
    const float inv_dx2 = 1.0f / (dx * dx);
    const float lf = (float)lptr[0];
    const float cfl = 0.5f * lf * (lf + 1.0f);

    float lmin = 3.0e38f, lmax = -3.0e38f;
    for (int i = threadIdx.x; i < NX; i += blockDim.x) {
        const float x = xc[i];
        const float cent = (x > 1e-10f) ? (cfl / (x * x)) : 0.0f;
        const float d = inv_dx2 + V_ext[i] + V_H[i] + V_xc[i] + cent;
        diag[i] = d;
        lmin = fminf(lmin, d);
        lmax = fmaxf(lmax, d);
    }
    smin[threadIdx.x] = lmin;
    smax[threadIdx.x] = lmax;
    __syncthreads();
    for (int s = blockDim.x >> 1; s > 0; s >>= 1) {
        if (threadIdx.x < s) {
            smin[threadIdx.x] = fminf(smin[threadIdx.x], smin[threadIdx.x + s]);
            smax[threadIdx.x] = fmaxf(smax[threadIdx.x], smax[threadIdx.x + s]);
        }
        __syncthreads();
    }
    if (threadIdx.x == 0) {
        const float ac = 0.5f * inv_dx2;          // |c|
        bounds[0] = smin[0] - 2.0f * ac - 1.0f;   // Gershgorin lower
        bounds[1] = smax[0] + 2.0f * ac + 1.0f;   // Gershgorin upper
    }
}

// ---------------------------------------------------------------------------
// Kernel 2: k-th eigenvalue by bisection with Sturm sign counts.
// One thread per eigenvalue. Diagonal staged in LDS (16 KB) so the
// 48 x 4096 recurrence steps per thread never leave the WGP.
// ---------------------------------------------------------------------------
__global__ void ks_bisect(const float* __restrict__ diag,
                          const float* __restrict__ bounds,
                          const float* __restrict__ dxp,
                          float* __restrict__ eigvals) {
    __shared__ float sd[NX];
    for (int i = threadIdx.x; i < NX; i += blockDim.x) sd[i] = diag[i];
    __syncthreads();

    const int k = blockIdx.x * blockDim.x + threadIdx.x;
    const float dx = dxp[0];
    const float c  = -0.5f / (dx * dx);
    const float cc = c * c;

    float lo = bounds[0];
    float hi = bounds[1];

    for (int it = 0; it < 48; ++it) {
        const float x = 0.5f * (lo + hi);
        int cnt = 0;
        float q = sd[0] - x;
        cnt += (q < 0.0f);
        for (int i = 1; i < NX; ++i) {
            if (fabsf(q) < 1e-30f) q = (q < 0.0f) ? -1e-30f : 1e-30f;
            q = (sd[i] - x) - cc / q;
            cnt += (q < 0.0f);
        }
        if (cnt <= k) lo = x; else hi = x;
    }
    eigvals[k] = 0.5f * (lo + hi);
}

// ---------------------------------------------------------------------------
// Kernel 3: eigenvector via 2 steps of inverse iteration on (T - lambda I).
// One thread per eigenvector; all global traffic is column-major so that
// consecutive threads touch consecutive addresses (fully coalesced).
// Pivot array u lives in d_ws (N floats per eigenvector); the rhs/solution
// lives in-place in the output eigenvector column.
// ---------------------------------------------------------------------------
__global__ void ks_invit(const float* __restrict__ diag,
                         const float* __restrict__ dxp,
                         const float* __restrict__ eigvals,
                         float* __restrict__ evecs,   // NX x NX row-major
                         float* __restrict__ uws) {   // NX x NX pivots
    __shared__ float sd[NX];
    for (int i = threadIdx.x; i < NX; i += blockDim.x) sd[i] = diag[i];
    __syncthreads();

    const int j = blockIdx.x * blockDim.x + threadIdx.x;
    const float dx  = dxp[0];
    const float c   = -0.5f / (dx * dx);
    const float lam = eigvals[j];

    // ---- iteration 1: rhs = ones -------------------------------------
    float u = sd[0] - lam;
    if (fabsf(u) < 1e-12f) u = (u < 0.0f) ? -1e-12f : 1e-12f;
    uws[j] = u;
    float y = 1.0f;
    evecs[j] = y;
    for (int i = 1; i < NX; ++i) {
        const float li = c / u;
        u = (sd[i] - lam) - li * c;
        if (fabsf(u) < 1e-12f) u = (u < 0.0f) ? -1e-12f : 1e-12f;
        uws[(size_t)i * NX + j] = u;
        y = 1.0f - li * y;
        evecs[(size_t)i * NX + j] = y;
    }
    float v = evecs[(size_t)(NX - 1) * NX + j] / uws[(size_t)(NX - 1) * NX + j];
    evecs[(size_t)(NX - 1) * NX + j] = v;
    float mx = fabsf(v);
    for (int i = NX - 2; i >= 0; --i) {
        v = (evecs[(size_t)i * NX + j] - c * v) / uws[(size_t)i * NX + j];
        evecs[(size_t)i * NX + j] = v;
        mx = fmaxf(mx, fabsf(v));
    }
    const float invmx = 1.0f / fmaxf(mx, 1e-30f);

    // ---- iteration 2: rhs = v / ||v||_inf (in place) ------------------
    u = sd[0] - lam;
    if (fabsf(u) < 1e-12f) u = (u < 0.0f) ? -1e-12f : 1e-12f;
    uws[j] = u;
    y = evecs[j] * invmx;
    evecs[j] = y;
    for (int i = 1; i < NX; ++i) {
        const float li = c / u;
        u = (sd[i] - lam) - li * c;
        if (fabsf(u) < 1e-12f) u = (u < 0.0f) ? -1e-12f : 1e-12f;
        uws[(size_t)i * NX + j] = u;
        y = evecs[(size_t)i * NX + j] * invmx - li * y;
        evecs[(size_t)i * NX + j] = y;
    }
    v = evecs[(size_t)(NX - 1) * NX + j] / uws[(size_t)(NX - 1) * NX + j];
    evecs[(size_t)(NX - 1) * NX + j] = v;
    for (int i = NX - 2; i >= 0; --i) {
        v = (evecs[(size_t)i * NX + j] - c * v) / uws[(size_t)i * NX + j];
        evecs[(size_t)i * NX + j] = v;
    }
}

// ---------------------------------------------------------------------------
// Kernel 4: vol-weighted panel Gram via V_WMMA_F32_16X16X4_F32, then
// normalize each eigenvector by rsqrt(diag(G)) — the reference's
// norm under the radial volume measure.  One wave32 per 16-column panel
// (EXEC all ones, as WMMA requires).
//   G(m,n) = sum_i vol[i] * V[i, c0+m] * V[i, c0+n]
//   A(m,k) = vol[i_k] * V[i_k, c0+m]   (16x4, f32, 2 VGPRs)
//   B(k,n) =           V[i_k, c0+n]    (4x16, f32, 2 VGPRs)
// A layout (ISA 7.12.2): lanes 0-15 hold K=0,1; lanes 16-31 hold K=2,3.
// C/D layout: VGPR r -> M=r (lanes 0-15) / M=r+8 (lanes 16-31), N=lane%16.
// ---------------------------------------------------------------------------
__global__ void ks_wmma_normalize(const float* __restrict__ vol,
                                  float* __restrict__ evecs) {
    __shared__ float G[16 * 16];
    __shared__ float invn[16];

    const int lane = threadIdx.x;      // 0..31, full wave
    const int half = lane >> 4;        // 0 or 1
    const int m    = lane & 15;        // row (A) / col (B) index — same lane map
    const int c0   = blockIdx.x * 16;  // panel base column

    v8f acc = {};
    const float* colbase = evecs + c0 + m;

    for (int i0 = 0; i0 < NX; i0 += 4) {
        const int r0 = i0 + 2 * half;          // K slots for this half-wave
        const float x0 = colbase[(size_t)r0 * NX];
        const float x1 = colbase[(size_t)(r0 + 1) * NX];
        v2f b; b.x = x0;            b.y = x1;
        v2f a; a.x = x0 * vol[r0];  a.y = x1 * vol[r0 + 1];
        acc = __builtin_amdgcn_wmma_f32_16x16x4_f32(
                  /*neg_a=*/false, a, /*neg_b=*/false, b,
                  /*c_mod=*/(short)0, acc, /*reuse_a=*/false, /*reuse_b=*/false);
    }

    // spill Gram tile to LDS per C/D layout
    for (int r = 0; r < 8; ++r) {
        const int M = r + half * 8;
        G[M * 16 + m] = acc[r];
    }
    __syncthreads();
    if (lane < 16) invn[lane] = rsqrtf(fmaxf(G[lane * 16 + lane], 1e-30f));
    __syncthreads();

    // scale the panel: half-wave 0 does even rows, half-wave 1 odd rows
    const float s = invn[m];
    for (int i = half; i < NX; i += 2) {
        evecs[(size_t)i * NX + c0 + m] *= s;
    }
}

// ---------------------------------------------------------------------------
// Host-side launch
// ---------------------------------------------------------------------------
extern "C" void kernel_launch(void* const* d_in, const int* in_sizes, int n_in,
                              void* d_out, int out_size, void* d_ws, size_t ws_size,
                              hipStream_t stream) {
    const int*   l     = (const int*)d_in[0];
    const float* V_ext = (const float*)d_in[1];
    const float* V_H   = (const float*)d_in[2];
    const float* V_xc  = (const float*)d_in[3];
    const float* xc    = (const float*)d_in[4];
    const float* vol   = (const float*)d_in[5];
    const float* dx    = (const float*)d_in[6];

    float* out     = (float*)d_out;
    float* eigvals = out;            // [NX]
    float* evecs   = out + NX;       // [NX * NX] row-major, col j = eigvec j

    float* wsf    = (float*)d_ws;
    float* diag   = wsf;             // [NX]
    float* bounds = wsf + NX;        // [2] (padded to 16)
    float* uws    = wsf + NX + 16;   // [NX * NX] pivot scratch

    ks_build_diag<<<1, 1024, 0, stream>>>(l, V_ext, V_H, V_xc, xc, dx, diag, bounds);
    ks_bisect<<<NX / 256, 256, 0, stream>>>(diag, bounds, dx, eigvals);
    ks_invit<<<NX / 256, 256, 0, stream>>>(diag, dx, eigvals, evecs, uws);
    ks_wmma_normalize<<<NX / 16, 32, 0, stream>>>(vol, evecs);
}